// Encoder_77395310674290
// MI455X (gfx1250) — compile-verified
//
#include <hip/hip_runtime.h>

// ============================================================================
// Capsule encoder for MI455X (gfx1250, wave32).
// Design: fuse conv-votes + dynamic routing per site (kills 134MB intermediate),
// chunk class-caps votes over batch so u-scratch (26MB) + W (52MB) stay inside
// the 192MB L2. Class votes use V_WMMA_F32_16X16X4_F32 (full fp32 precision).
// ============================================================================

#define BB    64
#define LL    512
#define KK    64
#define LNN   128
#define PREVV 5120
#define NCLSS 10
#define CHUNK 8
#define NCHUNK 8

typedef __attribute__((ext_vector_type(2))) float v2f;
typedef __attribute__((ext_vector_type(8))) float v8f;

// ---------------------------------------------------------------- conv1 ----
// x0[b,k,l] = c1b[k] + sum_g X[b,l+g-4] * c1w[k,0,g]
__global__ void k_conv1(const float* __restrict__ X, const float* __restrict__ w,
                        const float* __restrict__ bias, float* __restrict__ out) {
  int idx = blockIdx.x * blockDim.x + threadIdx.x;
  if (idx >= BB * KK * LL) return;
  int l = idx & (LL - 1);
  int k = (idx >> 9) & (KK - 1);
  int b = idx >> 15;
  float acc = bias[k];
  const float* xr = X + (size_t)b * LL;
  #pragma unroll
  for (int g = 0; g < 9; ++g) {
    int p = l + g - 4;
    float xv = (p >= 0 && p < LL) ? xr[p] : 0.f;
    acc = fmaf(xv, w[k * 9 + g], acc);
  }
  out[idx] = acc;
}

// ------------------------------------------------------------- A1 conv ----
// a1[b,o,l] = A1_b[o] + sum_{i<64,g<9} x0[b,i,l+g-4]*A1_w[o,i,g]
__global__ void k_convA1(const float* __restrict__ x0, const float* __restrict__ w,
                         const float* __restrict__ bias, float* __restrict__ out) {
  int idx = blockIdx.x * blockDim.x + threadIdx.x;
  if (idx >= BB * 64 * LL) return;
  int l = idx & (LL - 1);
  int o = (idx >> 9) & 63;
  int b = idx >> 15;
  float acc = bias[o];
  for (int i = 0; i < 64; ++i) {
    const float* xr = x0 + ((size_t)b * 64 + i) * LL;
    const float* wr = w + (o * 64 + i) * 9;
    #pragma unroll
    for (int g = 0; g < 9; ++g) {
      int p = l + g - 4;
      if (p >= 0 && p < LL) acc = fmaf(xr[p], wr[g], acc);
    }
  }
  out[idx] = acc;
}

// ---------------------------------------------------- squash A (over AP=8) ----
// in-place: element (b,l,p,a) = a1[b, p*8+a, l]
__global__ void k_squashA(float* __restrict__ a1) {
  int idx = blockIdx.x * blockDim.x + threadIdx.x;
  if (idx >= BB * 8 * LL) return;
  int l = idx & (LL - 1);
  int p = (idx >> 9) & 7;
  int b = idx >> 12;
  size_t base = ((size_t)b * 64 + p * 8) * LL + l;
  float vals[8];
  float sq = 0.f;
  #pragma unroll
  for (int a = 0; a < 8; ++a) { vals[a] = a1[base + (size_t)a * LL]; sq = fmaf(vals[a], vals[a], sq); }
  float scale = sq / (1.f + sq) * rsqrtf(sq + 1e-8f);
  #pragma unroll
  for (int a = 0; a < 8; ++a) a1[base + (size_t)a * LL] = vals[a] * scale;
}

// -------------------------- fused A2 conv2d + routing (M=8,C=8,D=16) ----
// one block (128 thr) per site (b,l); votes V[p][oc] computed in LDS.
__global__ void __launch_bounds__(128) k_cellA(const float* __restrict__ a1s,
    const float* __restrict__ w2, const float* __restrict__ bias2,
    const float* __restrict__ alpha_p, float* __restrict__ prev) {
  int site = blockIdx.x;
  int l = site & (LL - 1);
  int b = site >> 9;
  int t = threadIdx.x;
  __shared__ float sIn[3 * 64];
  __shared__ float V[8 * 128];
  __shared__ float blog[64];
  __shared__ float cmat[64];
  __shared__ float sv[128];
  __shared__ float vv[128];
  for (int j = t; j < 192; j += 128) {
    int dh = j >> 6, ch = j & 63;
    int lp = l + dh - 1;
    sIn[j] = (lp >= 0 && lp < LL) ? a1s[((size_t)b * 64 + ch) * LL + lp] : 0.f;
  }
  if (t < 64) blog[t] = 0.f;
  __syncthreads();
  {
    float bb = bias2[t];
    for (int p = 0; p < 8; ++p) {
      float acc = bb;
      #pragma unroll
      for (int dh = 0; dh < 3; ++dh)
        #pragma unroll
        for (int dw = 0; dw < 8; ++dw)
          acc = fmaf(sIn[dh * 64 + p * 8 + dw], w2[(t * 3 + dh) * 8 + dw], acc);
      V[p * 128 + t] = acc;
    }
  }
  __syncthreads();
  int c = t >> 4, d = t & 15;
  for (int r = 0; r < 3; ++r) {
    if (t < 64) {
      int m = t >> 3;
      float mx = blog[m * 8];
      for (int j = 1; j < 8; ++j) mx = fmaxf(mx, blog[m * 8 + j]);
      float den = 0.f;
      for (int j = 0; j < 8; ++j) den += __expf(blog[m * 8 + j] - mx);
      cmat[t] = __expf(blog[t] - mx) / den;
    }
    __syncthreads();
    float s = 0.f;
    for (int m = 0; m < 8; ++m) s = fmaf(cmat[m * 8 + c], V[m * 128 + t], s);
    sv[t] = s;
    __syncthreads();
    float sq = 0.f;
    for (int j = 0; j < 16; ++j) { float x = sv[c * 16 + j]; sq = fmaf(x, x, sq); }
    vv[t] = sq / (1.f + sq) * s * rsqrtf(sq + 1e-8f);
    __syncthreads();
    if (r < 2) {
      if (t < 64) {
        int m = t >> 3, cc = t & 7;
        float a = 0.f;
        for (int j = 0; j < 16; ++j) a = fmaf(V[m * 128 + cc * 16 + j], vv[cc * 16 + j], a);
        blog[t] += a;
      }
      __syncthreads();
    }
  }
  prev[((size_t)b * PREVV + l * 8 + c) * 16 + d] = alpha_p[0] * vv[t];
}

// ------------------------------------------------------------- BL (1x1) ----
__global__ void k_convBL(const float* __restrict__ x0, const float* __restrict__ w,
                         const float* __restrict__ bias, float* __restrict__ out) {
  int idx = blockIdx.x * blockDim.x + threadIdx.x;
  if (idx >= BB * 32 * LL) return;
  int l = idx & (LL - 1);
  int o = (idx >> 9) & 31;
  int b = idx >> 14;
  float acc = bias[o];
  const float* wr = w + o * 64;
  for (int i = 0; i < 64; ++i)
    acc = fmaf(x0[((size_t)b * 64 + i) * LL + l], wr[i], acc);
  out[idx] = acc;
}

// ------------------------------------------------------------- B1 conv ----
__global__ void k_convB1(const float* __restrict__ xb, const float* __restrict__ w,
                         const float* __restrict__ bias, float* __restrict__ out) {
  int idx = blockIdx.x * blockDim.x + threadIdx.x;
  if (idx >= BB * 256 * LL) return;
  int l = idx & (LL - 1);
  int o = (idx >> 9) & 255;
  int b = idx >> 17;
  float acc = bias[o];
  for (int i = 0; i < 32; ++i) {
    const float* xr = xb + ((size_t)b * 32 + i) * LL;
    const float* wr = w + (o * 32 + i) * 9;
    #pragma unroll
    for (int g = 0; g < 9; ++g) {
      int p = l + g - 4;
      if (p >= 0 && p < LL) acc = fmaf(xr[p], wr[g], acc);
    }
  }
  out[idx] = acc;
}

// ---------------------------------------------- squash B (over 256 ch) ----
// in-place on xb1 [b,ch,l]; one block per site (b,l), thread=ch
__global__ void __launch_bounds__(256) k_squashB(float* __restrict__ xb1) {
  int site = blockIdx.x;
  int l = site & (LL - 1);
  int b = site >> 9;
  int ch = threadIdx.x;
  size_t idx = ((size_t)b * 256 + ch) * LL + l;
  float v = xb1[idx];
  __shared__ float red[256];
  red[ch] = v * v;
  __syncthreads();
  for (int off = 128; off > 0; off >>= 1) {
    if (ch < off) red[ch] += red[ch + off];
    __syncthreads();
  }
  float sq = red[0];
  xb1[idx] = v * (sq / (1.f + sq) * rsqrtf(sq + 1e-8f));
}

// -------------------------- fused B2 conv2d + routing (M=4,C=8,D=16) ----
__global__ void __launch_bounds__(128) k_cellB(const float* __restrict__ xb1s,
    const float* __restrict__ w2, const float* __restrict__ bias2,
    const float* __restrict__ beta_p, float* __restrict__ prev) {
  int site = blockIdx.x;
  int ln = site & (LNN - 1);
  int b = site >> 7;
  int t = threadIdx.x;
  __shared__ float sIn[3 * 1024];
  __shared__ float V[4 * 128];
  __shared__ float blog[32];
  __shared__ float cmat[32];
  __shared__ float sv[128];
  __shared__ float vv[128];
  for (int j = t; j < 3072; j += 128) {
    int dh = j >> 10, jj = j & 1023;
    int n = jj >> 8, ch = jj & 255;
    int lnp = ln + dh - 1;
    sIn[j] = (lnp >= 0 && lnp < LNN) ? xb1s[((size_t)b * 256 + ch) * LL + lnp * 4 + n] : 0.f;
  }
  if (t < 32) blog[t] = 0.f;
  __syncthreads();
  {
    float bb = bias2[t];
    for (int n = 0; n < 4; ++n) {
      float acc = bb;
      for (int dh = 0; dh < 3; ++dh) {
        const float* wr = w2 + (t * 3 + dh) * 256;
        const float* ir = sIn + dh * 1024 + n * 256;
        for (int dw = 0; dw < 256; ++dw) acc = fmaf(ir[dw], wr[dw], acc);
      }
      V[n * 128 + t] = acc;
    }
  }
  __syncthreads();
  int c = t >> 4, d = t & 15;
  for (int r = 0; r < 3; ++r) {
    if (t < 32) {
      int m = t >> 3;
      float mx = blog[m * 8];
      for (int j = 1; j < 8; ++j) mx = fmaxf(mx, blog[m * 8 + j]);
      float den = 0.f;
      for (int j = 0; j < 8; ++j) den += __expf(blog[m * 8 + j] - mx);
      cmat[t] = __expf(blog[t] - mx) / den;
    }
    __syncthreads();
    float s = 0.f;
    for (int m = 0; m < 4; ++m) s = fmaf(cmat[m * 8 + c], V[m * 128 + t], s);
    sv[t] = s;
    __syncthreads();
    float sq = 0.f;
    for (int j = 0; j < 16; ++j) { float x = sv[c * 16 + j]; sq = fmaf(x, x, sq); }
    vv[t] = sq / (1.f + sq) * s * rsqrtf(sq + 1e-8f);
    __syncthreads();
    if (r < 2) {
      if (t < 32) {
        int m = t >> 3, cc = t & 7;
        float a = 0.f;
        for (int j = 0; j < 16; ++j) a = fmaf(V[m * 128 + cc * 16 + j], vv[cc * 16 + j], a);
        blog[t] += a;
      }
      __syncthreads();
    }
  }
  prev[((size_t)b * PREVV + 4096 + ln * 8 + c) * 16 + d] = beta_p[0] * vv[t];
}

// ------------------------------------------------- squash prev (D=16) ----
__global__ void k_squashPrev(float* __restrict__ prev) {
  int idx = blockIdx.x * blockDim.x + threadIdx.x;
  if (idx >= BB * PREVV) return;
  float* p = prev + (size_t)idx * 16;
  float vals[16];
  float sq = 0.f;
  #pragma unroll
  for (int j = 0; j < 16; ++j) { vals[j] = p[j]; sq = fmaf(vals[j], vals[j], sq); }
  float scale = sq / (1.f + sq) * rsqrtf(sq + 1e-8f);
  #pragma unroll
  for (int j = 0; j < 16; ++j) p[j] = vals[j] * scale;
}

// ----------------------------------------------------------- zero fill ----
__global__ void k_zero(float* __restrict__ p, int n) {
  int i = blockIdx.x * blockDim.x + threadIdx.x;
  if (i < n) p[i] = 0.f;
}

// ----------------------- class votes via V_WMMA_F32_16X16X4_F32 ----------
// u[m,c, row=b_local(8), e(16)] = sum_d xs[b0+b,m,d] * W[m,c,d,e]
// One wave per (m,c): D(16x16) = A(16x4 x4 steps) x B(4x16).
// A layout (ISA 7.12.2, f32 16x4): lane holds M=lane&15; K = 4k + 2*(lane>>4) + j.
// B layout mirrors with N=lane&15. D: VGPR j -> row j + 8*(lane>>4), col lane&15.
__global__ void __launch_bounds__(256) k_votes(const float* __restrict__ xs,
    const float* __restrict__ Wc, float* __restrict__ u, int b0) {
  int wave = (blockIdx.x * blockDim.x + threadIdx.x) >> 5;
  int lane = threadIdx.x & 31;
  if (wave >= PREVV * NCLSS) return;  // grid sized exactly; never taken
  int m = wave / NCLSS, c = wave % NCLSS;
  int hi = lane >> 4, lo = lane & 15;
  int beff = b0 + lo;
  if (beff > BB - 1) beff = BB - 1;   // rows >= CHUNK are don't-care
  const float* xrow = xs + ((size_t)beff * PREVV + m) * 16;
  const float* wrow = Wc + (size_t)(m * NCLSS + c) * 256;
  v8f acc = {};
  #pragma unroll
  for (int k = 0; k < 4; ++k) {
    int d0 = k * 4 + hi * 2;
    v2f a;  a.x = xrow[d0];             a.y = xrow[d0 + 1];
    v2f bm; bm.x = wrow[d0 * 16 + lo];  bm.y = wrow[(d0 + 1) * 16 + lo];
    acc = __builtin_amdgcn_wmma_f32_16x16x4_f32(false, a, false, bm,
                                                (short)0, acc, false, false);
  }
  // store only the CHUNK(=8) valid rows: hi==0 covers rows 0..7
  if (hi == 0) {
    float* ut = u + (size_t)(m * NCLSS + c) * (CHUNK * 16);
    #pragma unroll
    for (int j = 0; j < 8; ++j) ut[j * 16 + lo] = acc[j];
  }
}

// ---------------------------- class routing: s + squash(v) per (b,c) ----
__global__ void __launch_bounds__(256) k_sv(const float* __restrict__ u,
    const float* __restrict__ logit, float* __restrict__ vbuf,
    float* __restrict__ out, int b0, int write_out) {
  int bl = blockIdx.x / NCLSS;
  int c  = blockIdx.x % NCLSS;
  int t = threadIdx.x;
  int e = t & 15, m0 = t >> 4;
  float s = 0.f;
  for (int m = m0; m < PREVV; m += 16) {
    const float* lr = logit + ((size_t)bl * PREVV + m) * NCLSS;
    float mx = lr[0];
    #pragma unroll
    for (int j = 1; j < NCLSS; ++j) mx = fmaxf(mx, lr[j]);
    float den = 0.f;
    #pragma unroll
    for (int j = 0; j < NCLSS; ++j) den += __expf(lr[j] - mx);
    float w = __expf(lr[c] - mx) / den;
    s = fmaf(w, u[(size_t)(m * NCLSS + c) * (CHUNK * 16) + bl * 16 + e], s);
  }
  __shared__ float red[256];
  red[t] = s;
  __syncthreads();
  for (int off = 128; off >= 16; off >>= 1) {
    if (t < off) red[t] += red[t + off];
    __syncthreads();
  }
  if (t < 16) {
    float sq = 0.f;
    for (int j = 0; j < 16; ++j) { float x = red[j]; sq = fmaf(x, x, sq); }
    float v = sq / (1.f + sq) * red[t] * rsqrtf(sq + 1e-8f);
    vbuf[(bl * NCLSS + c) * 16 + t] = v;
    if (write_out) out[((size_t)(b0 + bl) * NCLSS + c) * 16 + t] = v;
  }
}

// ------------------------------- class routing: agreement -> logits ----
__global__ void k_upd(const float* __restrict__ u, const float* __restrict__ vbuf,
                      float* __restrict__ logit) {
  int idx = blockIdx.x * blockDim.x + threadIdx.x;
  if (idx >= CHUNK * PREVV * NCLSS) return;
  int bl = idx / (PREVV * NCLSS);
  int rem = idx % (PREVV * NCLSS);
  int m = rem / NCLSS, c = rem % NCLSS;
  const float* ur = u + (size_t)(m * NCLSS + c) * (CHUNK * 16) + bl * 16;
  const float* vr = vbuf + (bl * NCLSS + c) * 16;
  float a = 0.f;
  #pragma unroll
  for (int j = 0; j < 16; ++j) a = fmaf(ur[j], vr[j], a);
  logit[((size_t)bl * PREVV + m) * NCLSS + c] += a;
}

// ============================================================================
extern "C" void kernel_launch(void* const* d_in, const int* in_sizes, int n_in,
                              void* d_out, int out_size, void* d_ws, size_t ws_size,
                              hipStream_t stream) {
  (void)in_sizes; (void)n_in; (void)out_size;
  const float* X    = (const float*)d_in[0];
  const float* c1w  = (const float*)d_in[1];
  const float* c1b  = (const float*)d_in[2];
  const float* A1w  = (const float*)d_in[3];
  const float* A1b  = (const float*)d_in[4];
  const float* A2w  = (const float*)d_in[5];
  const float* A2b  = (const float*)d_in[6];
  const float* BLw  = (const float*)d_in[7];
  const float* BLb  = (const float*)d_in[8];
  const float* B1w  = (const float*)d_in[9];
  const float* B1b  = (const float*)d_in[10];
  const float* B2w  = (const float*)d_in[11];
  const float* B2b  = (const float*)d_in[12];
  const float* Wcls = (const float*)d_in[13];
  const float* alpha = (const float*)d_in[14];
  const float* beta  = (const float*)d_in[15];
  float* out = (float*)d_out;

  // ---- workspace layout (floats) ----
  float* ws = (float*)d_ws;
  size_t o = 0;
  float* x0    = ws + o; o += (size_t)BB * 64 * LL;            // 2,097,152
  float* big   = ws + o; o += (size_t)BB * 256 * LL;           // 8,388,608  (a1 then xb1)
  float* xb    = ws + o; o += (size_t)BB * 32 * LL;            // 1,048,576
  float* prev  = ws + o; o += (size_t)BB * PREVV * 16;         // 5,242,880
  float* u     = ws + o; o += (size_t)PREVV * NCLSS * CHUNK * 16; // 6,553,600
  float* logit = ws + o; o += (size_t)CHUNK * PREVV * NCLSS;   // 409,600
  float* vbuf  = ws + o; o += (size_t)CHUNK * NCLSS * 16;      // 1,280
  if (ws_size < o * sizeof(float)) return;  // ~95 MB required

  dim3 blk(256);
  // stem
  k_conv1<<<(BB * KK * LL + 255) / 256, blk, 0, stream>>>(X, c1w, c1b, x0);
  // cell A
  k_convA1<<<(BB * 64 * LL + 255) / 256, blk, 0, stream>>>(x0, A1w, A1b, big);
  k_squashA<<<(BB * 8 * LL + 255) / 256, blk, 0, stream>>>(big);
  k_cellA<<<BB * LL, 128, 0, stream>>>(big, A2w, A2b, alpha, prev);
  // cell B
  k_convBL<<<(BB * 32 * LL + 255) / 256, blk, 0, stream>>>(x0, BLw, BLb, xb);
  k_convB1<<<(BB * 256 * LL + 255) / 256, blk, 0, stream>>>(xb, B1w, B1b, big);
  k_squashB<<<BB * LL, 256, 0, stream>>>(big);
  k_cellB<<<BB * LNN, 128, 0, stream>>>(big, B2w, B2b, beta, prev);
  // class caps
  k_squashPrev<<<(BB * PREVV + 255) / 256, blk, 0, stream>>>(prev);
  for (int ch = 0; ch < NCHUNK; ++ch) {
    int b0 = ch * CHUNK;
    k_zero<<<(CHUNK * PREVV * NCLSS + 255) / 256, blk, 0, stream>>>(logit, CHUNK * PREVV * NCLSS);
    k_votes<<<(PREVV * NCLSS * 32) / 256, blk, 0, stream>>>(prev, Wcls, u, b0);
    for (int r = 0; r < 3; ++r) {
      k_sv<<<CHUNK * NCLSS, 256, 0, stream>>>(u, logit, vbuf, out, b0, (r == 2) ? 1 : 0);
      if (r < 2)
        k_upd<<<(CHUNK * PREVV * NCLSS + 255) / 256, blk, 0, stream>>>(u, vbuf, logit);
    }
  }
}